// GAT_77627238908070
// MI455X (gfx1250) — compile-verified
//
#include <hip/hip_runtime.h>
#include <hip/hip_bf16.h>

typedef __attribute__((ext_vector_type(2))) float v2f;
typedef __attribute__((ext_vector_type(8))) float v8f;

#define NIN   128
#define NHID  128
#define NOUT  2
#define SLOPE 0.2f

// ---------- helpers ----------
__device__ __forceinline__ void atomicAddF(float* p, float v) {
    (void)__hip_atomic_fetch_add(p, v, __ATOMIC_RELAXED, __HIP_MEMORY_SCOPE_AGENT);
}
// monotonic float <-> uint encoding for atomic max on floats
__device__ __forceinline__ unsigned fenc(float f) {
    int i = __float_as_int(f);
    return (i < 0) ? ~(unsigned)i : ((unsigned)i | 0x80000000u);
}
__device__ __forceinline__ float fdec(unsigned u) {
    int i = (u & 0x80000000u) ? (int)(u & 0x7fffffffu) : ~(int)u;
    return __int_as_float(i);
}
#define ENC_NEG_INF 0x007FFFFFu  // fenc(-inf)

// ---------- GEMM: H[n,128] = X[n,128] @ W[128,128]  (f32 WMMA 16x16x4) ----------
__device__ __forceinline__ v8f wmma_step(const float* __restrict__ xrow,
                                         const float* __restrict__ W,
                                         int col, int keff, v8f c) {
    // A 16x4 layout: lanes 0-15 hold K=k0,k0+1 ; lanes 16-31 hold K=k0+2,k0+3
    v2f a = *(const v2f*)(xrow + keff);
    v2f b;
    b.x = W[(size_t)keff * NHID + col];
    b.y = W[(size_t)(keff + 1) * NHID + col];
    return __builtin_amdgcn_wmma_f32_16x16x4_f32(false, a, false, b, (short)0, c, false, false);
}

__global__ __launch_bounds__(256) void gemm_wmma_f32(const float* __restrict__ X,
                                                     const float* __restrict__ W,
                                                     float* __restrict__ H) {
    const int wave = threadIdx.x >> 5;      // 0..7 -> N-tile (8*16 = 128 cols)
    const int lane = threadIdx.x & 31;
    const int row0 = blockIdx.x * 16;       // M-tile
    const int col0 = wave * 16;
    const int mlane = lane & 15;
    const int khi   = (lane >> 4) * 2;      // 0 or 2
    const int col   = col0 + mlane;
    const float* xrow = X + (size_t)(row0 + mlane) * NIN;

    v8f a0 = {}, a1 = {}, a2 = {}, a3 = {};
#pragma unroll
    for (int k0 = 0; k0 < NIN; k0 += 16) {
        a0 = wmma_step(xrow, W, col, k0 + 0  + khi, a0);
        a1 = wmma_step(xrow, W, col, k0 + 4  + khi, a1);
        a2 = wmma_step(xrow, W, col, k0 + 8  + khi, a2);
        a3 = wmma_step(xrow, W, col, k0 + 12 + khi, a3);
    }
    v8f acc = (a0 + a1) + (a2 + a3);

    // C/D layout: lanes 0-15 -> M = vgpr, lanes 16-31 -> M = 8 + vgpr
    const int mbase = row0 + ((lane >> 4) ? 8 : 0);
#pragma unroll
    for (int i = 0; i < 8; ++i)
        H[(size_t)(mbase + i) * NHID + col] = acc[i];
}

// ---------- per-node attention scores: ssrc = h.a_s, sdst = h.a_d ----------
__global__ __launch_bounds__(256) void node_scores(const float* __restrict__ H,
                                                   const float* __restrict__ as,
                                                   const float* __restrict__ ad,
                                                   float* __restrict__ ssrc,
                                                   float* __restrict__ sdst, int n) {
    const int node = blockIdx.x * 8 + (threadIdx.x >> 5);
    const int lane = threadIdx.x & 31;
    if (node >= n) return;
    const float4 h4 = ((const float4*)(H + (size_t)node * NHID))[lane];
    const float4 s4 = ((const float4*)as)[lane];
    const float4 d4 = ((const float4*)ad)[lane];
    float ss = h4.x * s4.x + h4.y * s4.y + h4.z * s4.z + h4.w * s4.w;
    float sd = h4.x * d4.x + h4.y * d4.y + h4.z * d4.z + h4.w * d4.w;
#pragma unroll
    for (int off = 16; off > 0; off >>= 1) {
        ss += __shfl_down(ss, off, 32);
        sd += __shfl_down(sd, off, 32);
    }
    if (lane == 0) { ssrc[node] = ss; sdst[node] = sd; }
}

// ---------- per-layer init: agg = bias, maxb = -inf, denom = 0 ----------
__global__ __launch_bounds__(256) void init_layer(float* __restrict__ agg,
                                                  const float* __restrict__ bias,
                                                  unsigned* __restrict__ maxb,
                                                  float* __restrict__ denom, int n) {
    const int t = blockIdx.x * 256 + threadIdx.x;
    if (t >= n * NHID) return;
    const int node = t >> 7, c = t & 127;
    agg[t] = bias[c];
    if (c == 0) { maxb[node] = ENC_NEG_INF; denom[node] = 0.f; }
}

// ---------- edge pass 1: leaky-relu logit + segment max ----------
__global__ __launch_bounds__(256) void edge_logit_max(const int* __restrict__ ei,
                                                      const float* __restrict__ ssrc,
                                                      const float* __restrict__ sdst,
                                                      float* __restrict__ ebuf,
                                                      unsigned* __restrict__ maxb,
                                                      int E, int n) {
    const int t = blockIdx.x * 256 + threadIdx.x;
    if (t >= E + n) return;
    int s, d;
    if (t < E) { s = ei[t]; d = ei[E + t]; } else { s = d = t - E; }  // self loops
    float e = ssrc[s] + sdst[d];
    e = (e > 0.f) ? e : SLOPE * e;
    ebuf[t] = e;
    (void)__hip_atomic_fetch_max(&maxb[d], fenc(e), __ATOMIC_RELAXED, __HIP_MEMORY_SCOPE_AGENT);
}

// ---------- edge pass 2: ex = exp(e - m[dst]); denom[dst] += ex ----------
__global__ __launch_bounds__(256) void edge_exp_sum(const int* __restrict__ ei,
                                                    float* __restrict__ ebuf,
                                                    const unsigned* __restrict__ maxb,
                                                    float* __restrict__ denom,
                                                    int E, int n) {
    const int t = blockIdx.x * 256 + threadIdx.x;
    if (t >= E + n) return;
    const int d = (t < E) ? ei[E + t] : (t - E);
    const float v = __expf(ebuf[t] - fdec(maxb[d]));
    ebuf[t] = v;
    atomicAddF(&denom[d], v);
}

// ---------- edge pass 3: agg[dst] += (ex/denom[dst]) * h[src]  (wave per edge) ----------
__global__ __launch_bounds__(256) void edge_scatter(const int* __restrict__ ei,
                                                    const float* __restrict__ ebuf,
                                                    const float* __restrict__ denom,
                                                    const float* __restrict__ H,
                                                    float* __restrict__ agg,
                                                    int E, int n) {
    const int w = blockIdx.x * 8 + (threadIdx.x >> 5);
    const int lane = threadIdx.x & 31;
    if (w >= E + n) return;
    int s, d;
    if (w < E) { s = ei[w]; d = ei[E + w]; } else { s = d = w - E; }
    const float alpha = ebuf[w] / denom[d];
    const float4 hv = ((const float4*)(H + (size_t)s * NHID))[lane];  // L2-resident gather
    float* ap = agg + (size_t)d * NHID + lane * 4;
    atomicAddF(ap + 0, alpha * hv.x);
    atomicAddF(ap + 1, alpha * hv.y);
    atomicAddF(ap + 2, alpha * hv.z);
    atomicAddF(ap + 3, alpha * hv.w);
}

// ---------- ReLU (in place) ----------
__global__ __launch_bounds__(256) void relu_inplace(float* __restrict__ a, int total) {
    const int t = blockIdx.x * 256 + threadIdx.x;
    if (t < total) a[t] = fmaxf(a[t], 0.f);
}

// ---------- classifier: out[n,2] = X @ Wc + bc ----------
__global__ __launch_bounds__(256) void classify(const float* __restrict__ X,
                                                const float* __restrict__ Wc,
                                                const float* __restrict__ bc,
                                                float* __restrict__ out, int n) {
    __shared__ float wc[NHID * NOUT];
    if (threadIdx.x < NHID * NOUT) wc[threadIdx.x] = Wc[threadIdx.x];
    __syncthreads();
    const int node = blockIdx.x * 256 + threadIdx.x;
    if (node >= n) return;
    const float* xr = X + (size_t)node * NHID;
    float s0 = bc[0], s1 = bc[1];
#pragma unroll 4
    for (int k = 0; k < NHID; ++k) {
        const float xv = xr[k];
        s0 += xv * wc[2 * k];
        s1 += xv * wc[2 * k + 1];
    }
    out[2 * node]     = s0;
    out[2 * node + 1] = s1;
}

extern "C" void kernel_launch(void* const* d_in, const int* in_sizes, int n_in,
                              void* d_out, int out_size, void* d_ws, size_t ws_size,
                              hipStream_t stream) {
    const float* x  = (const float*)d_in[0];
    const int*   ei = (const int*)d_in[1];
    const int n = in_sizes[0] / NIN;       // 100000
    const int E = in_sizes[1] / 2;         // 1600000
    const int tot = E + n;

    const float* W[3]  = { (const float*)d_in[2], (const float*)d_in[6],  (const float*)d_in[10] };
    const float* AS[3] = { (const float*)d_in[3], (const float*)d_in[7],  (const float*)d_in[11] };
    const float* AD[3] = { (const float*)d_in[4], (const float*)d_in[8],  (const float*)d_in[12] };
    const float* B[3]  = { (const float*)d_in[5], (const float*)d_in[9],  (const float*)d_in[13] };
    const float* Wc = (const float*)d_in[14];
    const float* bc = (const float*)d_in[15];

    // workspace layout (16B-aligned regions)
    char* ws = (char*)d_ws;
    size_t off = 0;
    auto alloc = [&](size_t bytes) { void* p = ws + off; off = (off + bytes + 15) & ~(size_t)15; return p; };
    float*    h     = (float*)alloc((size_t)n * NHID * sizeof(float));
    float*    agg   = (float*)alloc((size_t)n * NHID * sizeof(float));
    float*    ssrc  = (float*)alloc((size_t)n * sizeof(float));
    float*    sdst  = (float*)alloc((size_t)n * sizeof(float));
    float*    denom = (float*)alloc((size_t)n * sizeof(float));
    unsigned* maxb  = (unsigned*)alloc((size_t)n * sizeof(unsigned));
    float*    ebuf  = (float*)alloc((size_t)tot * sizeof(float));
    (void)ws_size;

    const int gridNode   = (n + 7) / 8;              // wave per node
    const int gridNC     = (n * NHID + 255) / 256;   // thread per (node,channel)
    const int gridEdge   = (tot + 255) / 256;        // thread per edge
    const int gridEdgeW  = (tot + 7) / 8;            // wave per edge

    const float* xin = x;
    for (int L = 0; L < 3; ++L) {
        gemm_wmma_f32 <<<n / 16,     256, 0, stream>>>(xin, W[L], h);
        node_scores   <<<gridNode,   256, 0, stream>>>(h, AS[L], AD[L], ssrc, sdst, n);
        init_layer    <<<gridNC,     256, 0, stream>>>(agg, B[L], maxb, denom, n);
        edge_logit_max<<<gridEdge,   256, 0, stream>>>(ei, ssrc, sdst, ebuf, maxb, E, n);
        edge_exp_sum  <<<gridEdge,   256, 0, stream>>>(ei, ebuf, maxb, denom, E, n);
        edge_scatter  <<<gridEdgeW,  256, 0, stream>>>(ei, ebuf, denom, h, agg, E, n);
        relu_inplace  <<<gridNC,     256, 0, stream>>>(agg, n * NHID);
        xin = agg;  // post-ReLU aggregate feeds next layer's GEMM
    }
    classify<<<(n + 255) / 256, 256, 0, stream>>>(agg, Wc, bc, (float*)d_out, n);
}